// Gfusedmax_76562087018943
// MI455X (gfx1250) — compile-verified
//
#include <hip/hip_runtime.h>

// Gfusedmax (graph fused lasso + sparsemax) for B=4096 rows, N=128.
// One workgroup (256 threads = 8 wave32) per batch row.
// z (128x128) and w=A (128x128) live entirely in VGPRs (8x8 tile per thread).
// Per-iteration row/col reductions: in-register partials -> WMMA f32 16x16x4
// combine (ones-matrix multiply), accumulator = colsum - rowsum.
// Clip implemented as a single v_med3_f32 per element.

typedef __attribute__((ext_vector_type(2))) float v2f;
typedef __attribute__((ext_vector_type(4))) float v4f;
typedef __attribute__((ext_vector_type(8))) float v8f;

#define NN    128
#define NITER 60
#define PST   17   // padded stride for partial matrices (gcd(17,64)=1 -> no LDS bank conflicts)

__launch_bounds__(256)
__global__ void gfusedmax_kernel(const float* __restrict__ x,
                                 const float* __restrict__ A,
                                 float* __restrict__ out) {
  __shared__ float Prow[NN * PST];   // partial rowsums: [row][col-group 0..15]
  __shared__ float Pcol[NN * PST];   // partial colsums: [col][row-group 0..15]
  __shared__ float xsh[NN];
  __shared__ float ybuf[NN];
  __shared__ float redv[NN];
  __shared__ float redc[NN];

  const int t    = threadIdx.x;
  const int b    = blockIdx.x;
  const int ti   = t >> 4;          // 0..15 row tile
  const int tj   = t & 15;          // 0..15 col tile
  const int r0   = ti << 3;
  const int c0   = tj << 3;
  const int lane = t & 31;
  const int wv   = t >> 5;          // 0..7 (wave id)

  if (t < NN) xsh[t] = x[(size_t)b * NN + t];

  // w = lam/gamma * A = A  -> registers (read-only bound for the clip)
  float w[8][8];
  {
    const v4f* Ab = (const v4f*)(A + (size_t)b * NN * NN);
    #pragma unroll
    for (int s = 0; s < 8; ++s) {
      int ro = (r0 + s) * (NN / 4) + (c0 >> 2);
      v4f q0 = Ab[ro];
      v4f q1 = Ab[ro + 1];
      w[s][0] = q0.x; w[s][1] = q0.y; w[s][2] = q0.z; w[s][3] = q0.w;
      w[s][4] = q1.x; w[s][5] = q1.y; w[s][6] = q1.z; w[s][7] = q1.w;
    }
  }

  float z[8][8];
  #pragma unroll
  for (int s = 0; s < 8; ++s)
    #pragma unroll
    for (int u = 0; u < 8; ++u) z[s][u] = 0.0f;

  float rp[8], cp[8];
  #pragma unroll
  for (int s = 0; s < 8; ++s) { rp[s] = 0.0f; cp[s] = 0.0f; }

  const float step   = 1.0f / (2.0f * NN);
  const v2f   bones  = { 1.0f,  1.0f };
  const v2f   bnones = {-1.0f, -1.0f};
  // WMMA A-operand (16x4 f32) gather indices:
  //   VGPR0: lanes0-15 K=k0+0, lanes16-31 K=k0+2 ; VGPR1: K=k0+1 / K=k0+3
  const int arow = (wv << 4) + (lane & 15);
  const int koff = (lane >> 4) << 1;

  for (int it = 0; it <= NITER; ++it) {
    // ---- phase 1: publish per-thread partial sums
    #pragma unroll
    for (int s = 0; s < 8; ++s) Prow[(r0 + s) * PST + tj] = rp[s];
    #pragma unroll
    for (int u = 0; u < 8; ++u) Pcol[(c0 + u) * PST + ti] = cp[u];
    __syncthreads();

    // ---- phase 2: WMMA combine, acc = colsum - rowsum (replicated across N-lanes)
    v8f acc = {};
    #pragma unroll
    for (int k0 = 0; k0 < 16; k0 += 4) {
      v2f a;
      a.x = Pcol[arow * PST + k0 + koff];
      a.y = Pcol[arow * PST + k0 + koff + 1];
      acc = __builtin_amdgcn_wmma_f32_16x16x4_f32(false, a, false, bones,
                                                  (short)0, acc, false, false);
    }
    #pragma unroll
    for (int k0 = 0; k0 < 16; k0 += 4) {
      v2f a;
      a.x = Prow[arow * PST + k0 + koff];
      a.y = Prow[arow * PST + k0 + koff + 1];
      acc = __builtin_amdgcn_wmma_f32_16x16x4_f32(false, a, false, bnones,
                                                  (short)0, acc, false, false);
    }
    // D layout: VGPR p, lanes0-15 -> M=p ; lanes16-31 -> M=p+8 (value replicated over N)
    if ((lane & 15) == 0) {
      int base = (wv << 4) + ((lane >> 4) << 3);
      #pragma unroll
      for (int p = 0; p < 8; ++p)
        ybuf[base + p] = xsh[base + p] + acc[p];
    }
    __syncthreads();

    if (it == NITER) break;

    // ---- phase 3: projected-gradient update of z, regenerate partials
    float yr[8], yc[8];
    #pragma unroll
    for (int s = 0; s < 8; ++s) yr[s] = ybuf[r0 + s];
    #pragma unroll
    for (int u = 0; u < 8; ++u) yc[u] = ybuf[c0 + u];
    #pragma unroll
    for (int s = 0; s < 8; ++s) { rp[s] = 0.0f; cp[s] = 0.0f; }
    #pragma unroll
    for (int s = 0; s < 8; ++s) {
      #pragma unroll
      for (int u = 0; u < 8; ++u) {
        float v   = z[s][u] + step * (yr[s] - yc[u]);
        float lim = w[s][u];
        v = __builtin_amdgcn_fmed3f(v, -lim, lim);   // clamp to [-lim, lim], lim >= 0
        z[s][u] = v;
        rp[s] += v;
        cp[u] += v;
      }
    }
  }

  // ---- sparsemax over ybuf (rank-based; reproduces sorted-mask formula)
  float yi = 0.0f;
  if (t < NN) {
    yi = ybuf[t];
    int   cnt = 0;
    float sgt = 0.0f;
    for (int j = 0; j < NN; ++j) {
      float yj = ybuf[j];
      bool  gt = (yj > yi) | ((yj == yi) & (j < t));  // strictly above in descending order
      if (gt) { cnt++; sgt += yj; }
    }
    float k  = (float)(cnt + 1);
    float cs = sgt + yi;
    bool  m  = (1.0f + k * yi > cs);
    redv[t] = m ? yi : 0.0f;
    redc[t] = m ? 1.0f : 0.0f;
  }
  __syncthreads();
  #pragma unroll
  for (int off = 64; off > 0; off >>= 1) {
    if (t < off) { redv[t] += redv[t + off]; redc[t] += redc[t + off]; }
    __syncthreads();
  }
  if (t < NN) {
    float tau = (redv[0] - 1.0f) / redc[0];
    out[(size_t)b * NN + t] = fmaxf(yi - tau, 0.0f);
  }
}

extern "C" void kernel_launch(void* const* d_in, const int* in_sizes, int n_in,
                              void* d_out, int out_size, void* d_ws, size_t ws_size,
                              hipStream_t stream) {
  const float* x = (const float*)d_in[0];   // [B, 128] f32
  const float* A = (const float*)d_in[1];   // [B, 128, 128] f32
  float* out = (float*)d_out;               // [B, 128] f32
  int Bn = in_sizes[0] / NN;
  gfusedmax_kernel<<<Bn, 256, 0, stream>>>(x, A, out);
}